// SemanticGuidedCompletionLoss_14293651161197
// MI455X (gfx1250) — compile-verified
//
#include <hip/hip_runtime.h>
#include <math.h>

typedef __attribute__((ext_vector_type(2))) float v2f;
typedef __attribute__((ext_vector_type(8))) float v8f;
typedef __attribute__((ext_vector_type(4))) unsigned int u32x4;
typedef __attribute__((ext_vector_type(4))) int i32x4v;
typedef __attribute__((ext_vector_type(8))) int i32x8v;

#define BETA_W  0.5f
#define F_THRESH2 0.0025f   /* 0.05^2 */

struct Accum {
  float rowsum[4];
  float colsum[4];
  float nll_sum;
  unsigned int valid_cnt;
  unsigned int rowcount[8];
  unsigned int colcount[8];
};

/* workspace layout (bytes) */
#define WS_COLMIN_OFF 0u
#define WS_TARGET_OFF (64u * 1024u)                      /* colmin: B*M uints = 64KB     */
#define WS_NNIDX_OFF  (WS_TARGET_OFF + 256u * 1024u)     /* target: B*Dq*f*3 f32 <=256KB */
#define WS_ACCUM_OFF  (WS_NNIDX_OFF + 96u * 1024u)       /* nn_idx: B*Nf ints <= 96KB    */

/* ---------------- init ---------------- */
__global__ void init_ws_kernel(unsigned int* __restrict__ colmin, int n,
                               Accum* __restrict__ acc, int initAccum) {
  int i = blockIdx.x * blockDim.x + threadIdx.x;
  if (i < n) colmin[i] = 0x7F800000u;   /* +inf bits */
  if (initAccum && i == 0) {
    unsigned int* p = (unsigned int*)acc;
    for (int k = 0; k < (int)(sizeof(Accum) / 4); ++k) p[k] = 0u;
  }
}

/* ---------------- tiled chamfer via V_WMMA_F32_16X16X4_F32 ----------------
 * One wave per 32-row slab of X (2 WMMA row tiles); sweeps columns of Y in
 * 2x16-wide tiles (4 WMMAs per iteration, B operands reused across row tiles).
 * A row i = (x0,x1,x2,1); B col j = (-2y0,-2y1,-2y2,|y|^2)  =>  C = |y|^2 - 2 x.y
 * dist = |x|^2 + C, clamped at 0.                                            */
template <bool TRACK>
__global__ __launch_bounds__(32)
void chamfer_wmma_kernel(const float* __restrict__ X, const float* __restrict__ Y,
                         int N, int M, int pair,
                         unsigned int* __restrict__ colmin,
                         Accum* __restrict__ acc,
                         int* __restrict__ nn_idx,   /* used when TRACK */
                         int do_count) {
  const int tilesPerB = N >> 5;          /* 32 rows per block */
  const int b    = blockIdx.x / tilesPerB;
  const int tile = blockIdx.x % tilesPerB;
  const int lane = threadIdx.x;
  const int hi   = lane >> 4;            /* K-half select */
  const int lj   = lane & 15;
  const int row0 = tile << 5;

  const float* __restrict__ Xb = X + (size_t)b * (size_t)N * 3u;
  const float* __restrict__ Yb = Y + (size_t)b * (size_t)M * 3u;

  /* A operands: two 16-row tiles; lane holds row (row0 + 16t + lj), K pair (2hi,2hi+1) of (x,1) */
  v2f a[2];
#pragma unroll
  for (int t = 0; t < 2; ++t) {
    const float* xp = Xb + (size_t)(row0 + 16 * t + lj) * 3u;
    float q0 = xp[0], q1 = xp[1], q2 = xp[2];
    a[t].x = hi ? q2 : q0;
    a[t].y = hi ? 1.0f : q1;
  }
  /* row norms for the 8 rows each accumulator covers (M = v + 8*hi) */
  float xx[2][8];
#pragma unroll
  for (int t = 0; t < 2; ++t)
#pragma unroll
    for (int v = 0; v < 8; ++v) {
      const float* xp = Xb + (size_t)(row0 + 16 * t + hi * 8 + v) * 3u;
      xx[t][v] = xp[0] * xp[0] + xp[1] * xp[1] + xp[2] * xp[2];
    }

  float rmin[2][8];
  int   ridx[2][8];
#pragma unroll
  for (int t = 0; t < 2; ++t)
#pragma unroll
    for (int v = 0; v < 8; ++v) { rmin[t][v] = 1e30f; ridx[t][v] = 0; }

  const int nChunk = M >> 4;
  for (int c = 0; c < nChunk; c += 2) {
    const int j0 = (c << 4) + lj;
    const int j1 = j0 + 16;
    if (c + 4 < nChunk) __builtin_prefetch(Yb + (size_t)(j0 + 64) * 3u, 0, 1);

    const float* g0 = Yb + (size_t)j0 * 3u;
    const float* g1 = Yb + (size_t)j1 * 3u;
    float y00 = g0[0], y01 = g0[1], y02 = g0[2];
    float y10 = g1[0], y11 = g1[1], y12 = g1[2];
    float yy0 = y00 * y00 + y01 * y01 + y02 * y02;
    float yy1 = y10 * y10 + y11 * y11 + y12 * y12;

    v2f b0, b1;
    b0.x = hi ? (-2.0f * y02) : (-2.0f * y00);
    b0.y = hi ? yy0           : (-2.0f * y01);
    b1.x = hi ? (-2.0f * y12) : (-2.0f * y10);
    b1.y = hi ? yy1           : (-2.0f * y11);

    v8f t00 = (v8f)0.0f, t01 = (v8f)0.0f, t10 = (v8f)0.0f, t11 = (v8f)0.0f;
    t00 = __builtin_amdgcn_wmma_f32_16x16x4_f32(false, a[0], false, b0, (short)0, t00, false, false);
    t10 = __builtin_amdgcn_wmma_f32_16x16x4_f32(false, a[1], false, b0, (short)0, t10, false, false);
    t01 = __builtin_amdgcn_wmma_f32_16x16x4_f32(false, a[0], false, b1, (short)0, t01, false, false);
    t11 = __builtin_amdgcn_wmma_f32_16x16x4_f32(false, a[1], false, b1, (short)0, t11, false, false);

    float cm0 = 1e30f, cm1 = 1e30f;
#pragma unroll
    for (int v = 0; v < 8; ++v) {
      float d00 = fmaxf(xx[0][v] + t00[v], 0.0f);
      float d10 = fmaxf(xx[1][v] + t10[v], 0.0f);
      float d01 = fmaxf(xx[0][v] + t01[v], 0.0f);
      float d11 = fmaxf(xx[1][v] + t11[v], 0.0f);
      if (TRACK) {
        if (d00 < rmin[0][v]) { rmin[0][v] = d00; ridx[0][v] = j0; }
        if (d01 < rmin[0][v]) { rmin[0][v] = d01; ridx[0][v] = j1; }
        if (d10 < rmin[1][v]) { rmin[1][v] = d10; ridx[1][v] = j0; }
        if (d11 < rmin[1][v]) { rmin[1][v] = d11; ridx[1][v] = j1; }
      } else {
        rmin[0][v] = fminf(rmin[0][v], fminf(d00, d01));
        rmin[1][v] = fminf(rmin[1][v], fminf(d10, d11));
      }
      cm0 = fminf(cm0, fminf(d00, d10));
      cm1 = fminf(cm1, fminf(d01, d11));
    }
    atomicMin(&colmin[(size_t)b * M + j0], __float_as_uint(cm0));
    atomicMin(&colmin[(size_t)b * M + j1], __float_as_uint(cm1));
  }

  /* cross-lane row-min reduction within each 16-lane half (prefer lower idx on tie) */
#pragma unroll
  for (int t = 0; t < 2; ++t) {
#pragma unroll
    for (int v = 0; v < 8; ++v) {
      float mv = rmin[t][v]; int iv = ridx[t][v];
      for (int mk = 1; mk < 16; mk <<= 1) {
        float om = __shfl_xor(mv, mk, 32);
        int   oi = __shfl_xor(iv, mk, 32);
        if (om < mv || (TRACK && om == mv && oi < iv)) { mv = om; iv = oi; }
      }
      rmin[t][v] = mv; ridx[t][v] = iv;
    }
  }

  if (lj == 0) {
    float s = 0.0f;
    unsigned int cnt = 0u;
#pragma unroll
    for (int t = 0; t < 2; ++t)
#pragma unroll
      for (int v = 0; v < 8; ++v) {
        int r = row0 + 16 * t + hi * 8 + v;
        float d = rmin[t][v];
        s += sqrtf(fmaxf(d, 1e-12f));
        if (do_count && d < F_THRESH2) cnt++;
        if (TRACK && nn_idx) nn_idx[(size_t)b * N + r] = ridx[t][v];
      }
    atomicAdd(&acc->rowsum[pair], s);
    if (do_count) atomicAdd(&acc->rowcount[b], cnt);
  }
}

/* ---------------- column-min reduction ---------------- */
__global__ __launch_bounds__(256)
void col_reduce_kernel(const unsigned int* __restrict__ colmin, int cols, int Bn,
                       int pair, Accum* __restrict__ acc, int do_count) {
  int i = blockIdx.x * blockDim.x + threadIdx.x;
  int total = Bn * cols;
  float s = 0.0f;
  int cnt = 0;
  int b = 0;
  if (i < total) {
    float d = __uint_as_float(colmin[i]);
    s = sqrtf(fmaxf(d, 1e-12f));
    b = i / cols;
    if (do_count && d < F_THRESH2) cnt = 1;
  }
  for (int mk = 16; mk; mk >>= 1) {
    s   += __shfl_down(s, mk, 32);
    cnt += __shfl_down(cnt, mk, 32);
  }
  if ((threadIdx.x & 31) == 0) {
    atomicAdd(&acc->colsum[pair], s);
    if (do_count && cnt > 0) atomicAdd(&acc->colcount[b], (unsigned int)cnt);
  }
}

/* ---------------- knn(factor) + gather ----------------
 * Stages the per-batch gt cloud (3*M dwords) into LDS with one Tensor Data
 * Mover DMA (tensor_load_to_lds, D# built per ISA 08 sect. 8.3/8.4), waits on
 * TENSORcnt, then runs the iterative argmin passes out of LDS.              */
__global__ __launch_bounds__(256)
void knn_gather_kernel(const float* __restrict__ dc, const float* __restrict__ gt,
                       int Dq, int M, int factor, float* __restrict__ target) {
  extern __shared__ float smem[];
  float* dist  = smem;          /* M floats           */
  float* ydata = smem + M;      /* 3*M floats (gt)    */
  __shared__ float sval[256];
  __shared__ int   sidx[256];
  const int b   = blockIdx.x / Dq;
  const int q   = blockIdx.x % Dq;
  const int tid = threadIdx.x;
  const float* __restrict__ gb = gt + (size_t)b * (size_t)M * 3u;

  if (tid < 32) {  /* wave 0 issues the TDM DMA: gt[b] -> LDS */
    unsigned long long ga = (unsigned long long)(uintptr_t)gb;
    unsigned nelem = 3u * (unsigned)M;                 /* dwords, fits tile_dim0 (16b) */
    unsigned ldsa  = (unsigned)(uintptr_t)ydata;       /* low 32 bits = LDS byte addr  */
    u32x4 g0;
    g0.x = 1u;                                          /* count=1, user mode          */
    g0.y = ldsa;                                        /* lds_addr                    */
    g0.z = (unsigned)(ga & 0xFFFFFFFFu);                /* global_addr[31:0]           */
    g0.w = (unsigned)((ga >> 32) & 0x1FFFFFFu) | (2u << 30);  /* addr[56:32] | type=2  */
    i32x8v g1;
    g1[0] = (int)(2u << 16);                            /* data_size=4B                */
    g1[1] = (int)((nelem & 0xFFFFu) << 16);             /* tensor_dim0 lo16            */
    g1[2] = (int)((nelem >> 16) | (1u << 16));          /* tensor_dim0 hi16|dim1 lo=1  */
    g1[3] = (int)(nelem << 16);                         /* dim1 hi=0 | tile_dim0       */
    g1[4] = 1;                                          /* tile_dim1=1, tile_dim2=0    */
    g1[5] = (int)nelem;                                 /* tensor_dim0_stride lo32     */
    g1[6] = 0;
    g1[7] = 0;
    i32x4v gz4 = {0, 0, 0, 0};
    i32x8v gz8 = {0, 0, 0, 0, 0, 0, 0, 0};
    /* amdgpu-toolchain (clang-23) 6-arg form */
    __builtin_amdgcn_tensor_load_to_lds(g0, g1, gz4, gz4, gz8, 0);
    __builtin_amdgcn_s_wait_tensorcnt(0);
  }
  __syncthreads();

  const float* __restrict__ p = dc + ((size_t)b * Dq + q) * 3u;
  float px = p[0], py = p[1], pz = p[2];
  for (int j = tid; j < M; j += blockDim.x) {
    float dx = px - ydata[j * 3 + 0];
    float dy = py - ydata[j * 3 + 1];
    float dz = pz - ydata[j * 3 + 2];
    dist[j] = dx * dx + dy * dy + dz * dz;
  }
  __syncthreads();

  for (int k = 0; k < factor; ++k) {
    float best = 1e30f;
    int   bi   = 0x7FFFFFFF;
    for (int j = tid; j < M; j += blockDim.x)
      if (dist[j] < best) { best = dist[j]; bi = j; }
    sval[tid] = best; sidx[tid] = bi;
    __syncthreads();
    for (int s = 128; s > 0; s >>= 1) {
      if (tid < s) {
        float ov = sval[tid + s]; int oi = sidx[tid + s];
        if (ov < sval[tid] || (ov == sval[tid] && oi < sidx[tid])) {
          sval[tid] = ov; sidx[tid] = oi;
        }
      }
      __syncthreads();
    }
    int sel = sidx[0];
    if (tid < 3)
      target[(((size_t)b * Dq + q) * factor + k) * 3u + tid] = ydata[sel * 3 + tid];
    if (tid == 0) dist[sel] = 1e30f;
    __syncthreads();
  }
}

/* ---------------- cross entropy over C logits ---------------- */
__global__ __launch_bounds__(256)
void ce_kernel(const float* __restrict__ logits, const int* __restrict__ gt_sem,
               const int* __restrict__ nn_idx, int Nf, int M, int C, int Bn,
               Accum* __restrict__ acc) {
  int i = blockIdx.x * blockDim.x + threadIdx.x;
  float nll = 0.0f;
  int valid = 0;
  if (i < Bn * Nf) {
    int b = i / Nf;
    int j = nn_idx[i];
    int label = gt_sem[(size_t)b * M + j];
    const float* lg = logits + (size_t)i * C;
    float mx = -1e30f;
    for (int c = 0; c < C; ++c) mx = fmaxf(mx, lg[c]);
    float se = 0.0f;
    for (int c = 0; c < C; ++c) se += __expf(lg[c] - mx);
    float lse = mx + __logf(se);
    if (label >= 0) { nll = lse - lg[label]; valid = 1; }
  }
  for (int mk = 16; mk; mk >>= 1) {
    nll   += __shfl_down(nll, mk, 32);
    valid += __shfl_down(valid, mk, 32);
  }
  if ((threadIdx.x & 31) == 0) {
    atomicAdd(&acc->nll_sum, nll);
    atomicAdd(&acc->valid_cnt, (unsigned int)valid);
  }
}

/* ---------------- final scalar combine ---------------- */
__global__ void final_kernel(const Accum* __restrict__ acc, const int* __restrict__ epoch_p,
                             int Bn, int Nc, int Nf, int M, int Dt,
                             float* __restrict__ out) {
  if (threadIdx.x != 0 || blockIdx.x != 0) return;
  float loss_coarse = 0.5f * (acc->rowsum[0] / (float)(Bn * Nc) + acc->colsum[0] / (float)(Bn * M));
  float loss_fine   = 0.5f * (acc->rowsum[1] / (float)(Bn * Nf) + acc->colsum[1] / (float)(Bn * M));
  float loss_den    = 0.5f * (acc->rowsum[2] / (float)(Bn * Dt) + acc->colsum[2] / (float)(Bn * Dt)) * BETA_W;

  float progress = (float)epoch_p[0] / 300.0f;
  float lam = (progress < 0.2f) ? 0.0f
            : (progress < 0.6f) ? 0.3f * (progress - 0.2f) / 0.4f
                                : 0.3f;
  float ce = acc->nll_sum / fmaxf((float)acc->valid_cnt, 1.0f);

  float total = loss_coarse + loss_fine + loss_den + lam * ce;

  float fs = 0.0f;
  for (int b = 0; b < Bn; ++b) {
    float pr = (float)acc->rowcount[b] / (float)Nf;
    float rc = (float)acc->colcount[b] / (float)M;
    fs += 2.0f * pr * rc / (pr + rc + 1e-8f);
  }
  fs /= (float)Bn;

  out[0] = total;
  out[1] = fs;
}

extern "C" void kernel_launch(void* const* d_in, const int* in_sizes, int n_in,
                              void* d_out, int out_size, void* d_ws, size_t ws_size,
                              hipStream_t stream) {
  const float* pred_coarse = (const float*)d_in[0];
  const float* pred_fine   = (const float*)d_in[1];
  const float* den_coarse  = (const float*)d_in[2];
  const float* den_fine    = (const float*)d_in[3];
  const float* sem_logits  = (const float*)d_in[4];
  const float* gt          = (const float*)d_in[5];
  const int*   gt_sem      = (const int*)d_in[6];
  const int*   epoch_p     = (const int*)d_in[8];

  const int B   = 2;
  const int Nc  = in_sizes[0] / (B * 3);
  const int Nf  = in_sizes[1] / (B * 3);
  const int Dq  = in_sizes[2] / (B * 3);
  const int Dtf = in_sizes[3] / (B * 3);           /* Dq * factor */
  const int M   = in_sizes[5] / (B * 3);
  const int C   = in_sizes[4] / (B * Nf);
  const int factor = Dtf / Dq;

  char* ws = (char*)d_ws;
  unsigned int* colmin = (unsigned int*)(ws + WS_COLMIN_OFF);
  float*        target = (float*)(ws + WS_TARGET_OFF);
  int*          nn_idx = (int*)(ws + WS_NNIDX_OFF);
  Accum*        acc    = (Accum*)(ws + WS_ACCUM_OFF);

  /* pair 0: pred_coarse vs gt */
  init_ws_kernel<<<(B * M + 255) / 256, 256, 0, stream>>>(colmin, B * M, acc, 1);
  chamfer_wmma_kernel<false><<<B * (Nc / 32), 32, 0, stream>>>(pred_coarse, gt, Nc, M, 0,
                                                               colmin, acc, nullptr, 0);
  col_reduce_kernel<<<(B * M + 255) / 256, 256, 0, stream>>>(colmin, M, B, 0, acc, 0);

  /* pair 1: pred_fine vs gt (nn_idx + f-score counts) */
  init_ws_kernel<<<(B * M + 255) / 256, 256, 0, stream>>>(colmin, B * M, acc, 0);
  chamfer_wmma_kernel<true><<<B * (Nf / 32), 32, 0, stream>>>(pred_fine, gt, Nf, M, 1,
                                                              colmin, acc, nn_idx, 1);
  col_reduce_kernel<<<(B * M + 255) / 256, 256, 0, stream>>>(colmin, M, B, 1, acc, 1);

  /* knn(factor) + gather -> denoised target (TDM-staged gt in LDS) */
  knn_gather_kernel<<<B * Dq, 256, (size_t)(4 * M) * sizeof(float), stream>>>(
      den_coarse, gt, Dq, M, factor, target);

  /* pair 2: denoised_fine vs gathered target */
  init_ws_kernel<<<(B * Dtf + 255) / 256, 256, 0, stream>>>(colmin, B * Dtf, acc, 0);
  chamfer_wmma_kernel<false><<<B * (Dtf / 32), 32, 0, stream>>>(den_fine, target, Dtf, Dtf, 2,
                                                                colmin, acc, nullptr, 0);
  col_reduce_kernel<<<(B * Dtf + 255) / 256, 256, 0, stream>>>(colmin, Dtf, B, 2, acc, 0);

  /* semantic CE */
  ce_kernel<<<(B * Nf + 255) / 256, 256, 0, stream>>>(sem_logits, gt_sem, nn_idx,
                                                      Nf, M, C, B, acc);

  /* combine */
  final_kernel<<<1, 1, 0, stream>>>(acc, epoch_p, B, Nc, Nf, M, Dtf, (float*)d_out);
}